// Neural_Network_60713657696427
// MI455X (gfx1250) — compile-verified
//
#include <hip/hip_runtime.h>

// Per-atom MLP (5 -> 64 -> 64 -> 1, tanh) over S structures.
// e[s][a] = W3_a . tanh(W2_a^T tanh(W1_a^T g[s,a] + b1_a) + b2_a) + b3_a
//
// Layout conventions (row-major, per setup_inputs):
//   g  [S][A][5],  W1 [A][5][64],  b1 [A][64]
//   W2 [A][64][64] (indexed [k_in][c_out]), b2 [A][64]
//   W3 [A][64][1], b3 [A][1],  out [S][A]
//
// Strategy: one block per (atom, S-chunk). Layer-2 GEMM via
// V_WMMA_F32_16X16X4_F32 (full fp32 precision), W2 pre-swizzled in LDS into
// the B-operand layout so each WMMA consumes one ds_load_b64 per lane
// (compiler pairs these into ds_load_2addr_stride64_b64).
// launch_bounds(256, 4) pins the register budget to <=256 VGPRs/lane so the
// allocator stops crossing the VGPR-MSB boundary (no s_set_vgpr_msb in the
// hot loop) while keeping >=4 waves/SIMD to hide HBM latency of streaming g.

typedef __attribute__((ext_vector_type(2))) float v2f;
typedef __attribute__((ext_vector_type(8))) float v8f;

#define S_TOT 4096
#define A_TOT 1024
#define H_DIM 64

__device__ __forceinline__ float fast_tanh(float x) {
    // tanh(x) = 1 - 2/(exp(2x)+1); exp->v_exp_f32, rcp->v_rcp_f32.
    float e = __expf(2.0f * x);
    return 1.0f - 2.0f * __builtin_amdgcn_rcpf(e + 1.0f);
}

__global__ __launch_bounds__(256, 4) void atomic_mlp_wmma_f32(
    const float* __restrict__ g,  const float* __restrict__ W1,
    const float* __restrict__ b1, const float* __restrict__ W2,
    const float* __restrict__ b2, const float* __restrict__ W3,
    const float* __restrict__ b3, float* __restrict__ out)
{
    const int atom = blockIdx.x;
    const int tid  = threadIdx.x;
    const int lane = tid & 31;
    const int wave = tid >> 5;      // 8 waves per block
    const int half = lane >> 4;     // wave32 half: selects K offset 0 / 2
    const int l15  = lane & 15;

    // W2 in WMMA B-operand layout: [j=K/4][n=N/16][lane], v2f = (K0, K0+1)
    __shared__ v2f  w2B[16 * 4 * 32];   // 16 KB
    __shared__ float w1s[5 * 64];
    __shared__ float b1s[64], b2s[64], w3s[64];
    __shared__ float b3s;

    const float* W2a = W2 + (size_t)atom * 64 * 64;
    for (int idx = tid; idx < 2048; idx += 256) {
        int ln = idx & 31;
        int n  = (idx >> 5) & 3;
        int j  = idx >> 7;
        int K0 = 4 * j + 2 * (ln >> 4);       // lanes 16-31 hold K+2,K+3
        int N  = 16 * n + (ln & 15);
        v2f v;
        v.x = W2a[(size_t)K0 * 64 + N];
        v.y = W2a[(size_t)(K0 + 1) * 64 + N];
        w2B[idx] = v;
    }
    const float* W1a = W1 + (size_t)atom * 5 * 64;
    for (int idx = tid; idx < 320; idx += 256) w1s[idx] = W1a[idx];
    if (tid < 64) {
        b1s[tid] = b1[(size_t)atom * 64 + tid];
        b2s[tid] = b2[(size_t)atom * 64 + tid];
        w3s[tid] = W3[(size_t)atom * 64 + tid];
    }
    if (tid == 0) b3s = b3[atom];
    __syncthreads();

    // Each block covers 32 strips of 16 S-rows; each wave takes 4 strips.
#pragma unroll 1
    for (int it = 0; it < 4; ++it) {
        const int strip = blockIdx.y * 32 + it * 8 + wave;
        const int s0    = strip * 16;

        // g row for this lane's matrix row (both halves read same rows).
        const float* grow = g + ((size_t)(s0 + l15) * A_TOT + atom) * 5;
        const float g0 = grow[0], g1 = grow[1], g2 = grow[2],
                    g3 = grow[3], g4 = grow[4];

        // ---- Layer 1 (K=5, VALU) directly into WMMA A-operand layout ----
        // lane<16: row=lane, K={4j,4j+1}; lane>=16: row=lane-16, K={4j+2,4j+3}
        v2f a[16];
        const int cb = 2 * half;
#pragma unroll
        for (int j = 0; j < 16; ++j) {
            int c0 = 4 * j + cb;
            int c1 = c0 + 1;
            float x0 = b1s[c0] + g0 * w1s[c0]       + g1 * w1s[64 + c0]
                               + g2 * w1s[128 + c0] + g3 * w1s[192 + c0]
                               + g4 * w1s[256 + c0];
            float x1 = b1s[c1] + g0 * w1s[c1]       + g1 * w1s[64 + c1]
                               + g2 * w1s[128 + c1] + g3 * w1s[192 + c1]
                               + g4 * w1s[256 + c1];
            a[j].x = fast_tanh(x0);
            a[j].y = fast_tanh(x1);
        }

        // ---- Layer 2 (WMMA f32 16x16x4) + fused layer 3 partials ----
        float acc[8];
#pragma unroll
        for (int v = 0; v < 8; ++v) acc[v] = 0.0f;

        // One live accumulator chain: keeps VGPRs bounded; resident waves
        // provide the ILP instead of in-wave chain interleaving.
#pragma unroll 1
        for (int n = 0; n < 4; ++n) {
            float bias = b2s[16 * n + l15];
            v8f C;
#pragma unroll
            for (int v = 0; v < 8; ++v) C[v] = bias;
#pragma unroll
            for (int j = 0; j < 16; ++j) {
                v2f b = w2B[j * 128 + n * 32 + lane];
                C = __builtin_amdgcn_wmma_f32_16x16x4_f32(
                        /*neg_a=*/false, a[j], /*neg_b=*/false, b,
                        /*c_mod=*/(short)0, C,
                        /*reuse_a=*/false, /*reuse_b=*/false);
            }
            float w3c = w3s[16 * n + l15];
#pragma unroll
            for (int v = 0; v < 8; ++v)
                acc[v] += fast_tanh(C[v]) * w3c;
        }

        // ---- Reduce the 16 N-lanes within each half (wave32 shfl) ----
#pragma unroll
        for (int m = 1; m < 16; m <<= 1) {
#pragma unroll
            for (int v = 0; v < 8; ++v)
                acc[v] += __shfl_xor(acc[v], m, 32);
        }

        // C layout: half0 VGPR v -> row v, half1 VGPR v -> row v+8.
        if (l15 == 0) {
            int rbase = s0 + half * 8;
#pragma unroll
            for (int v = 0; v < 8; ++v)
                out[(size_t)(rbase + v) * A_TOT + atom] = acc[v] + b3s;
        }
    }
}

extern "C" void kernel_launch(void* const* d_in, const int* in_sizes, int n_in,
                              void* d_out, int out_size, void* d_ws, size_t ws_size,
                              hipStream_t stream) {
    const float* g  = (const float*)d_in[0];
    const float* W1 = (const float*)d_in[1];
    const float* b1 = (const float*)d_in[2];
    const float* W2 = (const float*)d_in[3];
    const float* b2 = (const float*)d_in[4];
    const float* W3 = (const float*)d_in[5];
    const float* b3 = (const float*)d_in[6];
    float* out = (float*)d_out;

    dim3 grid(A_TOT, S_TOT / (16 * 32));  // (1024 atoms, 8 S-chunks)
    dim3 block(256);                      // 8 wave32 waves
    atomic_mlp_wmma_f32<<<grid, block, 0, stream>>>(g, W1, b1, W2, b2, W3, b3, out);
}